// NCC_gauss_730144441163
// MI455X (gfx1250) — compile-verified
//
#include <hip/hip_runtime.h>
#include <hip/hip_bf16.h>

// ---------------------------------------------------------------------------
// NCC (local Gaussian-window cross-correlation) for MI455X / gfx1250.
// Separable 9-tap Gaussian conv: pass X = LDS+scalar (contiguous axis, fused
// channel formation), passes Y/Z = V_WMMA_F32_16X16X4_F32 banded matmul with
// multi-tile waves (A-band built once, B chunks slide forward between tiles).
// Memory-bound (~1.2 GB streamed, mostly L2-resident on 192 MB L2).
// ---------------------------------------------------------------------------

typedef __attribute__((ext_vector_type(2))) float v2f;
typedef __attribute__((ext_vector_type(8))) float v8f;

#define W_ 160
#define H_ 192
#define D_ 160
#define VOL (160L * 192L * 160L)   // per-batch voxels (C == 1)

// 9-tap Gaussian, sigma = 2.0, normalized (precomputed in double, rounded).
__device__ __forceinline__ float gauss_tap(int t) {
  const float GW[9] = {
      0.027630551f, 0.066282245f, 0.123831500f, 0.180173815f, 0.204163685f,
      0.180173815f, 0.123831500f, 0.066282245f, 0.027630551f};
  // Select chain: folds to a constant when t is compile-time, and to a short
  // v_cndmask chain when t is lane-dependent (A-matrix band construction).
  float r = 0.0f;
#pragma unroll
  for (int k = 0; k < 9; ++k) r = (t == k) ? GW[k] : r;
  return r;
}

// ---------------------------------------------------------------------------
// Pass X: conv along contiguous W axis, fused with channel formation.
// One block per (d,h) row; 160 threads (5 wave32); inputs staged in LDS with
// a 4-element zero halo on each side. Writes 5 channel arrays:
//   CH[0]=G*I, CH[1]=G*J, CH[2]=G*I^2, CH[3]=G*J^2, CH[4]=G*(I*J)
// ---------------------------------------------------------------------------
__global__ void ncc_passx(const float* __restrict__ I, const float* __restrict__ J,
                          float* __restrict__ CH) {
  __shared__ float si[W_ + 8];
  __shared__ float sj[W_ + 8];
  const int w = threadIdx.x;           // 0..159
  const long row = blockIdx.x;         // 0..D_*H_-1
  const float* Ip = I + row * W_;
  const float* Jp = J + row * W_;
  si[w + 4] = Ip[w];
  sj[w + 4] = Jp[w];
  if (w < 4) {  // zero halo (reference uses zero 'same' padding)
    si[w] = 0.f; sj[w] = 0.f;
    si[w + W_ + 4] = 0.f; sj[w + W_ + 4] = 0.f;
  }
  __syncthreads();

  float aI = 0.f, aJ = 0.f, aII = 0.f, aJJ = 0.f, aIJ = 0.f;
#pragma unroll
  for (int t = 0; t < 9; ++t) {
    const float g = gauss_tap(t);            // compile-time constant here
    const float iv = si[w + t], jv = sj[w + t];
    aI  += g * iv;       aJ  += g * jv;
    aII += g * iv * iv;  aJJ += g * jv * jv; aIJ += g * iv * jv;
  }
  const long idx = row * W_ + w;
  CH[idx]            = aI;
  CH[VOL + idx]      = aJ;
  CH[2 * VOL + idx]  = aII;
  CH[3 * VOL + idx]  = aJJ;
  CH[4 * VOL + idx]  = aIJ;
}

// ---------------------------------------------------------------------------
// Strided-axis 9-tap conv as WMMA banded matmul, TILES output tiles per wave.
//   out[c0+p, w0+n] = sum_u Wband[p,u] * in[c0-4+u, w0+n],  Wband[p,u]=g[u-p]
// u=0..23 split into 6 K=4 chunks -> 6 x V_WMMA_F32_16X16X4_F32 per tile.
// A (16x4, constant band):  lane p(0-15): K=0,1 in a.x/a.y ; lane 16+p: K=2,3.
//   Built ONCE per wave, reused for all tiles.
// B (4x16, data):           lane n(0-15): rows K=0,1       ; lane 16+n: rows K=2,3.
//   Tile m chunks 4,5 (rows c0+12..19) == tile m+1 chunks 0,1 -> registers
//   slide forward; only 4 fresh chunks loaded per subsequent tile.
// D (16x16): VGPR r -> rows c0+r (lanes 0-15) / c0+8+r (lanes 16-31), col w0+n.
// One wave32 per block => EXEC all-ones at every WMMA (ISA requirement).
// ---------------------------------------------------------------------------
template <int TILES>
__global__ void ncc_conv_wmma(const float* __restrict__ in, float* __restrict__ out,
                              int Lc, int Sc, int So) {
  const int lane = threadIdx.x;        // 0..31
  const int p  = lane & 15;            // A row / B col / D col
  const int hi = lane >> 4;            // lane-half -> K split
  const int w0 = blockIdx.x * 16;
  const int c0 = blockIdx.y * (16 * TILES);
  const long obase = (long)blockIdx.z * So + w0;
  const float* inB = in + obase;
  float* outB = out + obase;

  // Constant Gaussian band, chunked: A[j] = Wband[:, 4j..4j+3]
  v2f A[6];
#pragma unroll
  for (int j = 0; j < 6; ++j) {
    const int u0 = 4 * j + 2 * hi;
    A[j].x = gauss_tap(u0 - p);
    A[j].y = gauss_tap(u0 + 1 - p);
  }

  // Chunk loader: rows rbase..rbase+3 (zero-padded outside [0,Lc)).
  auto load_chunk = [&](int rbase) -> v2f {
    const int r0 = rbase + 2 * hi;
    const int r1 = r0 + 1;
    v2f b;
    b.x = (r0 >= 0 && r0 < Lc) ? inB[(long)r0 * Sc + p] : 0.0f;
    b.y = (r1 >= 0 && r1 < Lc) ? inB[(long)r1 * Sc + p] : 0.0f;
    return b;
  };

  // Prime the 6-chunk window for the first tile (rows c0-4 .. c0+19).
  v2f B[6];
#pragma unroll
  for (int j = 0; j < 6; ++j) B[j] = load_chunk(c0 - 4 + 4 * j);

#pragma unroll
  for (int m = 0; m < TILES; ++m) {
    const int cm = c0 + 16 * m;

    v8f acc = {};
#pragma unroll
    for (int j = 0; j < 6; ++j) {
      acc = __builtin_amdgcn_wmma_f32_16x16x4_f32(
          /*neg_a=*/false, A[j], /*neg_b=*/false, B[j],
          /*c_mod=*/(short)0, acc, /*reuse_a=*/false, /*reuse_b=*/false);
    }

#pragma unroll
    for (int r = 0; r < 8; ++r) {
      const int row = cm + r + 8 * hi;
      outB[(long)row * Sc + p] = acc[r];
    }

    if (m + 1 < TILES) {
      // Slide window: next tile's chunks 0,1 are this tile's chunks 4,5.
      B[0] = B[4];
      B[1] = B[5];
      const int cn = cm + 16;
#pragma unroll
      for (int j = 2; j < 6; ++j) B[j] = load_chunk(cn - 4 + 4 * j);
    }
  }
}

// ---------------------------------------------------------------------------
// cc = sigma12^2 / (sigma1^2 * sigma2^2 + 1e-5), block-reduced, f64 atomic sum.
// ---------------------------------------------------------------------------
__global__ void ncc_cc_reduce(const float* __restrict__ CH, double* __restrict__ acc) {
  const long i = (long)blockIdx.x * blockDim.x + threadIdx.x;
  float cc = 0.f;
  if (i < VOL) {
    const float m1  = CH[i];
    const float m2  = CH[VOL + i];
    const float xII = CH[2 * VOL + i];
    const float xJJ = CH[3 * VOL + i];
    const float xIJ = CH[4 * VOL + i];
    const float s1  = xII - m1 * m1;
    const float s2  = xJJ - m2 * m2;
    const float s12 = xIJ - m1 * m2;
    cc = s12 * s12 / (s1 * s2 + 1e-5f);
  }
  __shared__ float red[256];
  red[threadIdx.x] = cc;
  __syncthreads();
  for (int s = 128; s > 0; s >>= 1) {
    if (threadIdx.x < s) red[threadIdx.x] += red[threadIdx.x + s];
    __syncthreads();
  }
  if (threadIdx.x == 0) atomicAdd(acc, (double)red[0]);  // global_atomic_add_f64
}

__global__ void ncc_zero_acc(double* acc) { *acc = 0.0; }

__global__ void ncc_finalize(const double* __restrict__ acc, float* __restrict__ out,
                             double inv_n) {
  *out = (float)(-(*acc) * inv_n);
}

// ---------------------------------------------------------------------------
// Host-side orchestration (graph-capture safe; all launches on `stream`).
// Scratch layout in d_ws: CH[5*VOL] f32 | tmp[VOL] f32 | acc f64  (~118 MB)
// ---------------------------------------------------------------------------
extern "C" void kernel_launch(void* const* d_in, const int* in_sizes, int n_in,
                              void* d_out, int out_size, void* d_ws, size_t ws_size,
                              hipStream_t stream) {
  const float* I = (const float*)d_in[0];  // y_true
  const float* J = (const float*)d_in[1];  // y_pred
  float* out = (float*)d_out;

  float* CH  = (float*)d_ws;               // 5 channel arrays, VOL each
  float* tmp = CH + 5 * VOL;               // y-pass ping buffer
  double* acc = (double*)((char*)d_ws + (size_t)6 * VOL * sizeof(float)); // 8B aligned

  ncc_zero_acc<<<1, 1, 0, stream>>>(acc);

  for (int b = 0; b < 2; ++b) {
    const float* Ib = I + (long)b * VOL;
    const float* Jb = J + (long)b * VOL;

    // Pass X (W axis, fused channel formation): one block per (d,h) row.
    ncc_passx<<<dim3(D_ * H_), dim3(W_), 0, stream>>>(Ib, Jb, CH);

    for (int ch = 0; ch < 5; ++ch) {
      float* a = CH + (long)ch * VOL;
      // Pass Y: conv along H (Lc=192, Sc=W=160, So=H*W=30720), other axis = D.
      // 4 tiles/wave: 192 = 3 groups * 64 rows.
      ncc_conv_wmma<4><<<dim3(W_ / 16, H_ / 64, D_), dim3(32), 0, stream>>>(
          a, tmp, H_, W_, H_ * W_);
      // Pass Z: conv along D (Lc=160, Sc=H*W=30720, So=W=160), other axis = H.
      // 2 tiles/wave: 160 = 5 groups * 32 rows.
      ncc_conv_wmma<2><<<dim3(W_ / 16, D_ / 32, H_), dim3(32), 0, stream>>>(
          tmp, a, D_, H_ * W_, W_);
    }

    ncc_cc_reduce<<<dim3((int)(VOL / 256)), dim3(256), 0, stream>>>(CH, acc);
  }

  ncc_finalize<<<1, 1, 0, stream>>>(acc, out, 1.0 / (double)(2 * VOL));
}